// MultiAttention_19112604467875
// MI455X (gfx1250) — compile-verified
//
#include <hip/hip_runtime.h>

typedef __attribute__((ext_vector_type(2))) float v2f;
typedef __attribute__((ext_vector_type(4))) float f4;
typedef __attribute__((ext_vector_type(8))) float v8f;

#define DIM 64
#define RR 1000
#define KN 10
#define BTN 96            // B*T
#define TILES_PER_BT 63   // ceil(1000/16)
#define SLICE (BTN * RR * DIM)  // 6,144,000 floats per tensor slice

static __device__ __forceinline__ v8f wmma_f32(v2f a, v2f b, v8f c) {
  // V_WMMA_F32_16X16X4_F32 : D = A(16x4) * B(4x16) + C(16x16), fp32
  return __builtin_amdgcn_wmma_f32_16x16x4_f32(
      /*neg_a=*/false, a, /*neg_b=*/false, b,
      /*c_mod=*/(short)0, c, /*reuse_a=*/false, /*reuse_b=*/false);
}

static __device__ __forceinline__ float sigmoidf(float x) {
  return 1.0f / (1.0f + __expf(-x));
}

// -------------------------------------------------------------------------
// Kernel A: KP = X @ W   (rows = 96000, D = 64), one 16-row tile per wave.
// -------------------------------------------------------------------------
__global__ __launch_bounds__(256) void proj64(const float* __restrict__ X,
                                              const float* __restrict__ W,
                                              float* __restrict__ Y) {
  __shared__ float Ws[DIM * DIM];        // 16 KB
  __shared__ float Xs[8][16 * DIM];      // 32 KB, one 16x64 tile per wave

  const int tid = threadIdx.x;
  const int lane = tid & 31;
  const int wid = tid >> 5;

  for (int i = tid; i < DIM * DIM / 4; i += 256)
    ((f4*)Ws)[i] = ((const f4*)W)[i];

  const int row0 = (blockIdx.x * 8 + wid) * 16;  // 750 blocks * 8 waves * 16 = 96000
  const f4* xsrc = (const f4*)(X + (size_t)row0 * DIM);
#pragma unroll
  for (int i = 0; i < 8; ++i)
    ((f4*)Xs[wid])[lane + 32 * i] = xsrc[lane + 32 * i];
  __syncthreads();

  const int m = lane & 15;
  const int half = lane >> 4;

  v8f acc[4] = {};
#pragma unroll
  for (int ks = 0; ks < 16; ++ks) {
    const int k0 = ks * 4 + half * 2;
    v2f a;
    a.x = Xs[wid][m * DIM + k0];
    a.y = Xs[wid][m * DIM + k0 + 1];
#pragma unroll
    for (int nt = 0; nt < 4; ++nt) {
      v2f b;
      b.x = Ws[k0 * DIM + nt * 16 + m];
      b.y = Ws[(k0 + 1) * DIM + nt * 16 + m];
      acc[nt] = wmma_f32(a, b, acc[nt]);
    }
  }

#pragma unroll
  for (int nt = 0; nt < 4; ++nt)
#pragma unroll
    for (int v = 0; v < 8; ++v)
      Y[(size_t)(row0 + half * 8 + v) * DIM + nt * 16 + m] = acc[nt][v];
}

// -------------------------------------------------------------------------
// Kernel B: full attention layer for one stream.
//   q = x@wq ; scores = q . KP[neigh] ; softmax ; out = q + attn.KP[neigh]
//   y = sigmoid(out @ wd)   (y may alias x: each row read only by own tile)
// One 16-row tile per wave, 8 waves per block, 756 blocks = 6048 tiles.
// -------------------------------------------------------------------------
__global__ __launch_bounds__(256) void attn64(const float* Xin,
                                              const float* __restrict__ KP,
                                              const int* __restrict__ IDX,
                                              const float* __restrict__ Wq,
                                              const float* __restrict__ Wd,
                                              float* Xout) {
  __shared__ float Wqs[DIM * DIM];       // 16 KB
  __shared__ float Wds[DIM * DIM];       // 16 KB
  __shared__ float Qs[8][16 * DIM];      // 32 KB  q tiles
  __shared__ float Ts[8][16 * DIM];      // 32 KB  x tile then out tile
  __shared__ float At[8][16 * KN];       // 5 KB   scores -> attn
  __shared__ int Ix[8][16 * KN];         // 5 KB   neighbor indices

  const int tid = threadIdx.x;
  const int lane = tid & 31;
  const int wid = tid >> 5;

  for (int i = tid; i < DIM * DIM / 4; i += 256) {
    ((f4*)Wqs)[i] = ((const f4*)Wq)[i];
    ((f4*)Wds)[i] = ((const f4*)Wd)[i];
  }

  const int tile = blockIdx.x * 8 + wid;       // 0..6047
  const int bt = tile / TILES_PER_BT;
  const int row0 = (tile - bt * TILES_PER_BT) * 16;
  const float* Xbt = Xin + (size_t)bt * RR * DIM;
  const float* KPbt = KP + (size_t)bt * RR * DIM;

  // Phase 1: stage x tile (row-clamped for the 8-row tail tile) + indices.
#pragma unroll
  for (int i = 0; i < 8; ++i) {
    const int e = lane + 32 * i;   // f4 index within tile (256 total)
    const int r = e >> 4;
    const int c = e & 15;
    int row = row0 + r;
    if (row > RR - 1) row = RR - 1;
    ((f4*)Ts[wid])[e] = ((const f4*)(Xbt + (size_t)row * DIM))[c];
  }
  for (int p = lane; p < 16 * KN; p += 32) {
    const int r = p / KN;
    const int j = p - r * KN;
    int row = row0 + r;
    if (row > RR - 1) row = RR - 1;
    Ix[wid][p] = IDX[row * KN + j];
  }
  __syncthreads();

  const int m = lane & 15;
  const int half = lane >> 4;

  // Phase 2: q = x @ Wq (WMMA), write to Qs.
  v8f accq[4] = {};
#pragma unroll
  for (int ks = 0; ks < 16; ++ks) {
    const int k0 = ks * 4 + half * 2;
    v2f a;
    a.x = Ts[wid][m * DIM + k0];
    a.y = Ts[wid][m * DIM + k0 + 1];
#pragma unroll
    for (int nt = 0; nt < 4; ++nt) {
      v2f b;
      b.x = Wqs[k0 * DIM + nt * 16 + m];
      b.y = Wqs[(k0 + 1) * DIM + nt * 16 + m];
      accq[nt] = wmma_f32(a, b, accq[nt]);
    }
  }
#pragma unroll
  for (int nt = 0; nt < 4; ++nt)
#pragma unroll
    for (int v = 0; v < 8; ++v)
      Qs[wid][(half * 8 + v) * DIM + nt * 16 + m] = accq[nt][v];
  __syncthreads();

  // Phase 3: scores[r][j] = q[r] . KP[idx[r][j]]  (160 dot64, 5 per lane)
  for (int p = lane; p < 16 * KN; p += 32) {
    const int r = p / KN;
    const f4* kr = (const f4*)(KPbt + (size_t)Ix[wid][p] * DIM);
    float s = 0.f;
#pragma unroll
    for (int d4 = 0; d4 < 16; ++d4) {
      const f4 kv = kr[d4];
      const f4 qv = ((const f4*)&Qs[wid][r * DIM])[d4];
      s += qv.x * kv.x + qv.y * kv.y + qv.z * kv.z + qv.w * kv.w;
    }
    At[wid][p] = s;
  }
  __syncthreads();

  // Phase 4: softmax over K=10 per row (lanes 0..15, one row each).
  if (lane < 16) {
    float* srow = &At[wid][lane * KN];
    float mx = srow[0];
#pragma unroll
    for (int j = 1; j < KN; ++j) mx = fmaxf(mx, srow[j]);
    float e[KN];
    float sum = 0.f;
#pragma unroll
    for (int j = 0; j < KN; ++j) {
      e[j] = __expf(srow[j] - mx);
      sum += e[j];
    }
    const float inv = 1.f / sum;
#pragma unroll
    for (int j = 0; j < KN; ++j) srow[j] = e[j] * inv;
  }
  __syncthreads();

  // Phase 5: out[r] = q[r] + sum_j attn[r][j] * KP[idx[r][j]]  -> Ts
  // lane handles columns d=lane and d=lane+32 (coalesced KP reads).
#pragma unroll
  for (int mm = 0; mm < 16; ++mm) {
    float o0 = Qs[wid][mm * DIM + lane];
    float o1 = Qs[wid][mm * DIM + lane + 32];
#pragma unroll
    for (int j = 0; j < KN; ++j) {
      const float aw = At[wid][mm * KN + j];
      const float* kr = KPbt + (size_t)Ix[wid][mm * KN + j] * DIM;
      o0 += aw * kr[lane];
      o1 += aw * kr[lane + 32];
    }
    Ts[wid][mm * DIM + lane] = o0;
    Ts[wid][mm * DIM + lane + 32] = o1;
  }
  __syncthreads();

  // Phase 6: y = sigmoid(out @ Wd) (WMMA), masked store for tail tile.
  v8f acco[4] = {};
#pragma unroll
  for (int ks = 0; ks < 16; ++ks) {
    const int k0 = ks * 4 + half * 2;
    v2f a;
    a.x = Ts[wid][m * DIM + k0];
    a.y = Ts[wid][m * DIM + k0 + 1];
#pragma unroll
    for (int nt = 0; nt < 4; ++nt) {
      v2f b;
      b.x = Wds[k0 * DIM + nt * 16 + m];
      b.y = Wds[(k0 + 1) * DIM + nt * 16 + m];
      acco[nt] = wmma_f32(a, b, acco[nt]);
    }
  }
  float* Yout = Xout + (size_t)bt * RR * DIM;
#pragma unroll
  for (int nt = 0; nt < 4; ++nt)
#pragma unroll
    for (int v = 0; v < 8; ++v) {
      const int rl = row0 + half * 8 + v;
      if (rl < RR)
        Yout[(size_t)rl * DIM + nt * 16 + m] = sigmoidf(acco[nt][v]);
    }
}

// -------------------------------------------------------------------------
// Final combine: out = sigmoid(xp*wp[d] + xr*wr[d] + xc*wc[d])
// -------------------------------------------------------------------------
__global__ __launch_bounds__(256) void combine3(const float* __restrict__ xp,
                                                const float* __restrict__ xr,
                                                const float* __restrict__ xc,
                                                const float* __restrict__ wp,
                                                const float* __restrict__ wr,
                                                const float* __restrict__ wc,
                                                float* __restrict__ out) {
  const size_t i = (size_t)blockIdx.x * 256 + threadIdx.x;
  if (i < (size_t)SLICE) {
    const int d = (int)(i & (DIM - 1));
    const float v = xp[i] * wp[d] + xr[i] * wr[d] + xc[i] * wc[d];
    out[i] = sigmoidf(v);
  }
}

extern "C" void kernel_launch(void* const* d_in, const int* in_sizes, int n_in,
                              void* d_out, int out_size, void* d_ws, size_t ws_size,
                              hipStream_t stream) {
  const float* all_data = (const float*)d_in[0];
  const int* idx[3] = {(const int*)d_in[1], (const int*)d_in[2], (const int*)d_in[3]};
  const float* wq[3] = {(const float*)d_in[4], (const float*)d_in[7], (const float*)d_in[10]};
  const float* wk[3] = {(const float*)d_in[5], (const float*)d_in[8], (const float*)d_in[11]};
  const float* wd[3] = {(const float*)d_in[6], (const float*)d_in[9], (const float*)d_in[12]};
  const float* wv[3] = {(const float*)d_in[13], (const float*)d_in[14], (const float*)d_in[15]};

  float* wsf = (float*)d_ws;
  float* xbuf[3] = {wsf, wsf + (size_t)SLICE, wsf + 2 * (size_t)SLICE};
  float* kp = (float*)d_out;  // d_out doubles as KP scratch; combine3 overwrites it last

  for (int sp = 0; sp < 2; ++sp) {
    for (int s = 0; s < 3; ++s) {
      const float* xin = (sp == 0) ? all_data : xbuf[s];
      // KP = xin @ wk  (project-then-gather: K x fewer FLOPs than reference)
      proj64<<<750, 256, 0, stream>>>(xin, wk[s] + sp * DIM * DIM, kp);
      // full attention layer; writes xbuf[s] (in place at sp==1, safe per-row)
      attn64<<<756, 256, 0, stream>>>(xin, kp, idx[s], wq[s] + sp * DIM * DIM,
                                      wd[s] + sp * DIM * DIM, xbuf[s]);
    }
  }
  combine3<<<(SLICE + 255) / 256, 256, 0, stream>>>(
      xbuf[0], xbuf[1], xbuf[2], wv[0], wv[1], wv[2], (float*)d_out);
}